// PNAAggregator_38663295598922
// MI455X (gfx1250) — compile-verified
//
#include <hip/hip_runtime.h>
#include <math.h>

// ---- problem constants (from reference) ----
#define N_EDGES   1600000
#define N_NODES   50000
#define EPSV      1e-5f
#define AVG_LOG   3.42951220f  // mean(log(deg+1)) for deg = {16,32,32,48}

// ---- workspace layout (float/uint element offsets) ----
#define OFF_CNT   0u
#define OFF_SUM   50048u
#define OFF_SSQ   (OFF_SUM + 6400000u)
#define OFF_MIN   (OFF_SSQ + 6400000u)   // uint-encoded float min
#define OFF_MAX   (OFF_MIN + 6400000u)   // uint-encoded float max
#define OFF_G     (OFF_MAX + 6400000u)   // 3 * 512 * 128 floats, fragment-swizzled
#define INIT_TOT  OFF_G
#define G_ELEMS   (3u * 512u * 128u)

typedef __attribute__((ext_vector_type(2))) float v2f;
typedef __attribute__((ext_vector_type(4))) float v4f;
typedef __attribute__((ext_vector_type(8))) float v8f;

// order-preserving float <-> uint encoding so uint atomicMin/Max == float min/max
__device__ __forceinline__ unsigned encf(float f) {
    unsigned b = __float_as_uint(f);
    return (b & 0x80000000u) ? ~b : (b | 0x80000000u);
}
__device__ __forceinline__ float decf(unsigned e) {
    unsigned b = (e & 0x80000000u) ? (e & 0x7FFFFFFFu) : ~e;
    return __uint_as_float(b);
}

// ---------------- kernel 1: init accumulators ----------------
__global__ void pna_init_ws(float* __restrict__ wf, unsigned* __restrict__ wu) {
    unsigned i = blockIdx.x * 256u + threadIdx.x;
    if (i >= INIT_TOT) return;
    if (i < OFF_MIN)       wf[i] = 0.0f;          // cnt, sum, ssq
    else if (i < OFF_MAX)  wu[i] = 0xFFFFFFFFu;   // min-enc -> +inf-like
    else                   wu[i] = 0u;            // max-enc -> -inf-like
}

// ---------------- kernel 2: build projection G in WMMA B-fragment order ----
// Math: G_s[k = t*128+p*32+f, col = b*32+l] = sum_i phm_rule[i,p,b] * W[i, s*128+t*32+f, l]
// Storage (fragment-swizzled): GF[((s*8+ct)*64 + kblk)*128 + lane*4 + pair*2 + j]
//   ct = col/16, ln = col%16, kstep = k/4, half = (k%4)/2, j = k%2,
//   lane = half*16+ln, kblk = kstep/2, pair = kstep%2.
// Each lane's 16B chunk = B fragments for K-steps (kblk*8) and (kblk*8+4).
__global__ void pna_build_G(const float* __restrict__ phm, const float* __restrict__ W,
                            float* __restrict__ GF) {
    unsigned id = blockIdx.x * 256u + threadIdx.x;
    if (id >= G_ELEMS) return;
    unsigned col   = id & 127u;
    unsigned k     = (id >> 7) & 511u;
    unsigned s     = id >> 16;            // 512*128 = 65536 per s
    unsigned t     = k >> 7;
    unsigned inner = k & 127u;
    unsigned p     = inner >> 5;
    unsigned f     = inner & 31u;
    unsigned b     = col >> 5;
    unsigned l     = col & 31u;
    unsigned kw    = s * 128u + t * 32u + f;   // row within W's K dim (384)
    float acc = 0.0f;
#pragma unroll
    for (int i = 0; i < 4; ++i)
        acc += phm[i * 16 + p * 4 + b] * W[((unsigned)i * 384u + kw) * 32u + l];

    // swizzled destination address
    unsigned ct    = col >> 4;
    unsigned ln    = col & 15u;
    unsigned kstep = k >> 2;
    unsigned krem  = k & 3u;
    unsigned half  = krem >> 1;
    unsigned j     = krem & 1u;
    unsigned lane  = half * 16u + ln;
    unsigned kblk  = kstep >> 1;
    unsigned pair  = kstep & 1u;
    unsigned dst = (((s * 8u + ct) * 64u + kblk) * 32u + lane) * 4u + pair * 2u + j;
    GF[dst] = acc;
}

// ---------------- kernel 3: edge scatter (one wave per edge) ----------------
__global__ void pna_scatter(const float4* __restrict__ x4, const int* __restrict__ idx,
                            float* __restrict__ wf, unsigned* __restrict__ wu) {
    unsigned gt   = blockIdx.x * 256u + threadIdx.x;
    unsigned e    = gt >> 5;
    unsigned lane = gt & 31u;
    if (e >= N_EDGES) return;

    // stream-ahead prefetch of x (global_prefetch_b8)
    if (e + 256u < N_EDGES)
        __builtin_prefetch(&x4[(size_t)(e + 256u) * 32u + lane], 0, 0);

    int node = idx[e];
    float4 v = x4[(size_t)e * 32u + lane];
    unsigned base = (unsigned)node * 128u + lane * 4u;

    float*    sum = wf + OFF_SUM;
    float*    ssq = wf + OFF_SSQ;
    unsigned* mn  = wu + OFF_MIN;
    unsigned* mx  = wu + OFF_MAX;

    float vals[4] = { v.x, v.y, v.z, v.w };
#pragma unroll
    for (int j = 0; j < 4; ++j) {
        float val = vals[j];
        atomicAdd(&sum[base + j], val);          // global_atomic_add_f32 (L2-resident)
        atomicAdd(&ssq[base + j], val * val);
        unsigned enc = encf(val);
        atomicMin(&mn[base + j], enc);           // global_atomic_min_u32
        atomicMax(&mx[base + j], enc);           // global_atomic_max_u32
    }
    if (lane == 0)
        atomicAdd(&wf[OFF_CNT + (unsigned)node], 1.0f);
}

// ---------------- kernel 4: fused stats + WMMA GEMM ----------------
// One block = 16 nodes. Phase A: stats tile S(16x512) -> LDS (padded stride 516
// so stride-512 A-fragment column reads hit 16 distinct banks). Phase B: 8 waves,
// each owns a 16-wide output column tile; per 8-K chunk: 2 LDS dual-loads (A),
// 3 coalesced global_load_b128 (B fragments), 6 v_wmma_f32_16x16x4_f32.
// Phase C: per-row degree-scaler combine + bias + store.
__global__ void __launch_bounds__(256)
pna_finalize(const float* __restrict__ wf, const unsigned* __restrict__ wu,
             const float* __restrict__ GF, const float* __restrict__ bias,
             float* __restrict__ out) {
    __shared__ float S[16][516];
    __shared__ float ls[16];   // log_deg / AVG_LOG
    __shared__ float asc[16];  // attenuation scale

    unsigned tid   = threadIdx.x;
    unsigned tile0 = blockIdx.x * 16u;

    const float*    sum = wf + OFF_SUM;
    const float*    ssq = wf + OFF_SSQ;
    const unsigned* mnE = wu + OFF_MIN;
    const unsigned* mxE = wu + OFF_MAX;

    // Phase A: build stats tile
    for (unsigned i = tid; i < 16u * 128u; i += 256u) {
        unsigned m = i >> 7, c = i & 127u;
        unsigned node = tile0 + m;
        float cnt  = wf[OFF_CNT + node];
        float safe = fmaxf(cnt, 1.0f);
        float s    = sum[node * 128u + c];
        float q    = ssq[node * 128u + c];
        float mean = s / safe;
        float var  = q / safe - mean * mean;
        float sd   = sqrtf(fmaxf(var, 0.0f) + EPSV);
        float mnv  = (cnt > 0.0f) ? decf(mnE[node * 128u + c]) : 0.0f;
        float mxv  = (cnt > 0.0f) ? decf(mxE[node * 128u + c]) : 0.0f;
        S[m][c]        = mean;
        S[m][128u + c] = mnv;
        S[m][256u + c] = mxv;
        S[m][384u + c] = sd;
    }
    if (tid < 16u) {
        float deg = wf[OFF_CNT + tile0 + tid];
        float ld  = logf(deg + 1.0f);
        ls[tid]  = ld / AVG_LOG;
        asc[tid] = (deg == 0.0f) ? 1.0f : (AVG_LOG / ld);
    }
    __syncthreads();

    // Phase B: WMMA over K = 512, three accumulators (base / amp / att)
    unsigned w    = tid >> 5;        // wave id 0..7 -> column tile
    unsigned lane = tid & 31u;
    unsigned half = lane >> 4;       // K half-split per A/B fragment layout
    unsigned ln   = lane & 15u;
    unsigned col  = w * 16u + ln;

    v8f a0 = {}, a1 = {}, a2 = {};
    // per-(s, col-tile) fragment panels: 64 kblks * 128 floats each
    const float* P0 = GF + ((0u * 8u + w) * 64u) * 128u + lane * 4u;
    const float* P1 = GF + ((1u * 8u + w) * 64u) * 128u + lane * 4u;
    const float* P2 = GF + ((2u * 8u + w) * 64u) * 128u + lane * 4u;

    for (unsigned kblk = 0; kblk < 64u; ++kblk) {
        unsigned k0 = kblk * 8u;
        unsigned kk = k0 + 2u * half;
        // A fragments for K-steps k0 and k0+4 (lanes 0-15: K=kk,kk+1; 16-31: +2)
        v2f A0 = *(const v2f*)&S[ln][kk];
        v2f A1 = *(const v2f*)&S[ln][kk + 4u];
        // B fragments: one b128 per matrix = fragments for both K-steps
        v4f B0 = *(const v4f*)&P0[kblk * 128u];
        v4f B1 = *(const v4f*)&P1[kblk * 128u];
        v4f B2 = *(const v4f*)&P2[kblk * 128u];
        v2f b0a; b0a.x = B0.x; b0a.y = B0.y;
        v2f b1a; b1a.x = B1.x; b1a.y = B1.y;
        v2f b2a; b2a.x = B2.x; b2a.y = B2.y;
        v2f b0b; b0b.x = B0.z; b0b.y = B0.w;
        v2f b1b; b1b.x = B1.z; b1b.y = B1.w;
        v2f b2b; b2b.x = B2.z; b2b.y = B2.w;
        a0 = __builtin_amdgcn_wmma_f32_16x16x4_f32(false, A0, false, b0a, (short)0, a0, false, false);
        a1 = __builtin_amdgcn_wmma_f32_16x16x4_f32(false, A0, false, b1a, (short)0, a1, false, false);
        a2 = __builtin_amdgcn_wmma_f32_16x16x4_f32(false, A0, false, b2a, (short)0, a2, false, false);
        a0 = __builtin_amdgcn_wmma_f32_16x16x4_f32(false, A1, false, b0b, (short)0, a0, false, false);
        a1 = __builtin_amdgcn_wmma_f32_16x16x4_f32(false, A1, false, b1b, (short)0, a1, false, false);
        a2 = __builtin_amdgcn_wmma_f32_16x16x4_f32(false, A1, false, b2b, (short)0, a2, false, false);
    }

    // Phase C: per-row scaler combine. C/D layout: VGPR v -> M = v + 8*(lane>=16), N = ln
    float bcol = bias[col];
#pragma unroll
    for (int v = 0; v < 8; ++v) {
        unsigned m = (unsigned)v + half * 8u;
        float r = a0[v] + ls[m] * a1[v] + asc[m] * a2[v] + bcol;
        out[(size_t)(tile0 + m) * 128u + col] = r;
    }
}

// ---------------- launch ----------------
extern "C" void kernel_launch(void* const* d_in, const int* in_sizes, int n_in,
                              void* d_out, int out_size, void* d_ws, size_t ws_size,
                              hipStream_t stream) {
    (void)in_sizes; (void)n_in; (void)out_size; (void)ws_size;
    const float* x    = (const float*)d_in[0];
    const int*   idx  = (const int*)d_in[1];
    const float* phm  = (const float*)d_in[2];
    const float* W    = (const float*)d_in[3];
    const float* bvec = (const float*)d_in[4];
    float*    out = (float*)d_out;
    float*    wf  = (float*)d_ws;
    unsigned* wu  = (unsigned*)d_ws;

    pna_init_ws<<<(INIT_TOT + 255u) / 256u, 256, 0, stream>>>(wf, wu);
    pna_build_G<<<(G_ELEMS + 255u) / 256u, 256, 0, stream>>>(phm, W, wf + OFF_G);
    {
        unsigned total = (unsigned)N_EDGES * 32u;
        pna_scatter<<<(total + 255u) / 256u, 256, 0, stream>>>((const float4*)x, idx, wf, wu);
    }
    pna_finalize<<<N_NODES / 16, 256, 0, stream>>>(wf, wu, wf + OFF_G, bvec, out);
}